// ResidualVQ_58428735094917
// MI455X (gfx1250) — compile-verified
//
#include <hip/hip_runtime.h>
#include <hip/hip_bf16.h>

typedef float v2f __attribute__((ext_vector_type(2)));
typedef float v8f __attribute__((ext_vector_type(8)));
typedef unsigned u32x4 __attribute__((ext_vector_type(4)));
typedef int i32x8 __attribute__((ext_vector_type(8)));
typedef int i32x4 __attribute__((ext_vector_type(4)));

#define N_TOK   65536          // 32*2048 tokens
#define DIM     256
#define KCODES  1024
#define NQ      4
#define ROWS_PER_BLOCK 128     // 8 waves * 16 rows
#define CB_PAD  260            // LDS row stride (floats): 256 data + 4 pad

// d_out element offsets (floats), reference return order
#define OFF_QOUT  0u
#define PLANE     16777216u                    // N_TOK*DIM
#define OFF_LOSS  16777216u
#define OFF_PERP  16777217u
#define OFF_IDX   16777218u
#define OFF_ALLQ  17039362u                    // OFF_IDX + NQ*N_TOK

#if defined(__has_builtin)
#if __has_builtin(__builtin_amdgcn_tensor_load_to_lds)
#define HAS_TDM 1
#endif
#endif

// ---------------------------------------------------------------------------
// Zero workspace accumulators (hist[NQ][KCODES] u32, then loss_acc[NQ] f32)
// ---------------------------------------------------------------------------
__global__ void rvq_init(unsigned* ws) {
  int i = blockIdx.x * blockDim.x + threadIdx.x;
  if (i < NQ * KCODES + NQ) ws[i] = 0u;
}

// ---------------------------------------------------------------------------
// Precompute ||c||^2 for all NQ*KCODES codes (one wave per code)
// ---------------------------------------------------------------------------
__global__ __launch_bounds__(256)
void rvq_csq(const float* __restrict__ cb, float* __restrict__ csq) {
  const int wave = threadIdx.x >> 5, lane = threadIdx.x & 31;
  const int code = blockIdx.x * 8 + wave;            // 0 .. NQ*KCODES-1
  const float* row = cb + (size_t)code * DIM + lane * 8;
  float4 a = ((const float4*)row)[0];
  float4 b = ((const float4*)row)[1];
  float s = a.x*a.x + a.y*a.y + a.z*a.z + a.w*a.w
          + b.x*b.x + b.y*b.y + b.z*b.z + b.w*b.w;
#pragma unroll
  for (int m = 16; m >= 1; m >>= 1) s += __shfl_xor(s, m, 32);
  if (lane == 0) csq[code] = s;
}

// ---------------------------------------------------------------------------
// TDM: async-load one 16x256 f32 codebook tile into LDS with a 2D D#.
// pad_interval=7 (pad after every 256 DWORDs), pad_amount=3 (+4 DWORDs)
// -> LDS rows land at stride CB_PAD=260 floats, conflict-free for b64 reads.
// 6-arg builtin form (amdgpu-toolchain / clang-23): groups 2,3 + trailing
// group are zero (2D tensor, no cluster multicast, no gather).
// ---------------------------------------------------------------------------
__device__ __forceinline__ void tdm_load_tile(const float* src, void* ldsDst) {
#ifdef HAS_TDM
  const unsigned lds = (unsigned)(unsigned long long)ldsDst; // flat lo32 == LDS offset
  const unsigned long long ga = (unsigned long long)src;
  u32x4 g0;
  g0.x = 1u;                                            // count=1 (valid user D#)
  g0.y = lds;                                           // lds_addr (bytes)
  g0.z = (unsigned)ga;                                  // global_addr[31:0]
  g0.w = (unsigned)((ga >> 32) & 0x01FFFFFFu)           // global_addr[56:32]
       | (2u << 30);                                    // type = 2 ("image")
  i32x8 g1;
  g1[0] = (2 << 16)      // data_size = 4 bytes
        | (1 << 20)      // pad_enable
        | (7 << 22)      // pad_interval: 256 DWORDs
        | (3 << 25);     // pad_amount: 4 DWORDs
  g1[1] = (int)(256u << 16);   // tensor_dim0 = 256 elements (lo16)
  g1[2] = (int)(16u << 16);    // tensor_dim0 hi=0 | tensor_dim1 = 16 rows (lo16)
  g1[3] = (int)(256u << 16);   // tensor_dim1 hi=0 | tile_dim0 = 256
  g1[4] = 16;                  // tile_dim1 = 16, tile_dim2 = 0
  g1[5] = 256;                 // tensor_dim0_stride = 256 elements (lo32)
  g1[6] = 0;
  g1[7] = 0;
  const i32x4 z4 = {0, 0, 0, 0};
  const i32x8 z8 = {0, 0, 0, 0, 0, 0, 0, 0};
  __builtin_amdgcn_tensor_load_to_lds(g0, g1, z4, z4, z8, 0);
#endif
}

// ---------------------------------------------------------------------------
// Fused distance-GEMM (f32 WMMA) + argmin + gather + histogram + loss partial
// grid: (N_TOK/128, NQ), block: 256 (8 waves)
// ---------------------------------------------------------------------------
__global__ __launch_bounds__(256)
void rvq_argmin(const float* __restrict__ x,
                const float* __restrict__ codebooks,
                const float* __restrict__ csqbuf,
                float* __restrict__ out,
                unsigned* __restrict__ hist,
                float* __restrict__ loss_acc) {
  __shared__ float s_cb[2][16 * CB_PAD];  // double-buffered code tiles
  __shared__ float s_xsq[8 * 16];
  __shared__ int   s_idx[8 * 16];
  __shared__ float s_red[256];

  const int tid    = threadIdx.x;
  const int wave   = tid >> 5;
  const int lane   = tid & 31;
  const int laneLo = lane & 15;
  const int half   = lane >> 4;
  const int q      = blockIdx.y;
  const size_t rowBase = (size_t)blockIdx.x * ROWS_PER_BLOCK;
  const float* cbq  = codebooks + (size_t)q * KCODES * DIM;
  const float* csqq = csqbuf + (size_t)q * KCODES;

#ifdef HAS_TDM
  if (wave == 0) tdm_load_tile(cbq, &s_cb[0][0]);       // overlap with A loads
#endif

  // ---- A fragments: 16x256 per wave, ISA 16x4 f32 A layout, in registers ----
  const size_t myRow = rowBase + (size_t)wave * 16 + laneLo;
  const float* xrow  = x + myRow * DIM;
  v2f a[64];
  float sa = 0.f;
#pragma unroll
  for (int i = 0; i < 64; ++i) {
    a[i] = *(const v2f*)(xrow + 4 * i + 2 * half);
    sa += a[i].x * a[i].x + a[i].y * a[i].y;
  }
  float xsqFull = sa + __shfl_xor(sa, 16, 32);          // combine K-halves
  if (half == 0) s_xsq[wave * 16 + laneLo] = xsqFull;

#ifdef HAS_TDM
  if (wave == 0) __builtin_amdgcn_s_wait_tensorcnt(0);
#else
  {  // fallback: cooperative load of tile 0
#pragma unroll
    for (int j = 0; j < 4; ++j) {
      int v4 = j * 256 + tid, row = v4 >> 6, col = (v4 & 63) * 4;
      *(float4*)&s_cb[0][row * CB_PAD + col] =
          *(const float4*)(cbq + (size_t)row * DIM + col);
    }
  }
#endif
  __syncthreads();

  float xsq_r[8];
#pragma unroll
  for (int r = 0; r < 8; ++r) xsq_r[r] = s_xsq[wave * 16 + r + 8 * half];

  float best[8];
  int   bidx[8];
#pragma unroll
  for (int r = 0; r < 8; ++r) { best[r] = 3.4e38f; bidx[r] = 0; }

  // ---- 64 tiles of 16 codes, TDM double-buffered ----
  for (int t = 0; t < KCODES / 16; ++t) {
    const int cur = t & 1;
    const int c0  = t * 16;
#ifdef HAS_TDM
    if (wave == 0 && t + 1 < KCODES / 16)
      tdm_load_tile(cbq + (size_t)(c0 + 16) * DIM, &s_cb[1 - cur][0]);
#endif
    const float csq = csqq[c0 + laneLo];                // precomputed ||c_n||^2

    v8f c = {};
    const float* bbase = &s_cb[cur][laneLo * CB_PAD + 2 * half];
#pragma unroll
    for (int i = 0; i < 64; ++i) {                      // pure ds_load + wmma
      v2f b = *(const v2f*)(bbase + 4 * i);
      c = __builtin_amdgcn_wmma_f32_16x16x4_f32(
              false, a[i], false, b, (short)0, c, false, false);
    }
    const int n = c0 + laneLo;
#pragma unroll
    for (int r = 0; r < 8; ++r) {
      float dist = xsq_r[r] - 2.0f * c[r] + csq;
      if (dist < best[r]) { best[r] = dist; bidx[r] = n; }  // first-min kept
    }

    if (t + 1 < KCODES / 16) {
#ifdef HAS_TDM
      if (wave == 0) __builtin_amdgcn_s_wait_tensorcnt(0);
      __syncthreads();
#else
      __syncthreads();
#pragma unroll
      for (int j = 0; j < 4; ++j) {
        int v4 = j * 256 + tid, row = v4 >> 6, col = (v4 & 63) * 4;
        *(float4*)&s_cb[1 - cur][row * CB_PAD + col] =
            *(const float4*)(cbq + (size_t)(c0 + 16 + row) * DIM + col);
      }
      __syncthreads();
#endif
    }
  }

  // ---- argmin across the 16 column-lanes of each half-wave ----
#pragma unroll
  for (int r = 0; r < 8; ++r) {
    float d = best[r];
    int   i = bidx[r];
#pragma unroll
    for (int m = 8; m >= 1; m >>= 1) {
      float od = __shfl_xor(d, m, 32);
      int   oi = __shfl_xor(i, m, 32);
      if (od < d || (od == d && oi < i)) { d = od; i = oi; }
    }
    if (laneLo == 0) s_idx[wave * 16 + r + 8 * half] = i;
  }
  __syncthreads();

  // ---- gather codes, emit all_quantized + indices + histogram + loss ----
  float acc = 0.f;
  const int dBase = lane * 8;
  for (int rr = 0; rr < 16; ++rr) {
    size_t m   = rowBase + (size_t)wave * 16 + rr;
    int    idx = s_idx[wave * 16 + rr];
    const float* crow = cbq + (size_t)idx * DIM;
    float4 qa = *(const float4*)(crow + dBase);
    float4 qb = *(const float4*)(crow + dBase + 4);
    float4 xa = *(const float4*)(x + m * DIM + dBase);
    float4 xb = *(const float4*)(x + m * DIM + dBase + 4);
    // all_quantized plane base is only 8B-aligned -> float2 stores
    float* dst = out + OFF_ALLQ + ((size_t)q * N_TOK + m) * DIM + dBase;
    ((float2*)dst)[0] = make_float2(qa.x, qa.y);
    ((float2*)dst)[1] = make_float2(qa.z, qa.w);
    ((float2*)dst)[2] = make_float2(qb.x, qb.y);
    ((float2*)dst)[3] = make_float2(qb.z, qb.w);
    float d0 = qa.x-xa.x, d1 = qa.y-xa.y, d2 = qa.z-xa.z, d3 = qa.w-xa.w;
    float d4 = qb.x-xb.x, d5 = qb.y-xb.y, d6 = qb.z-xb.z, d7 = qb.w-xb.w;
    acc += d0*d0 + d1*d1 + d2*d2 + d3*d3 + d4*d4 + d5*d5 + d6*d6 + d7*d7;
    if (lane == 0) {
      atomicAdd(&hist[q * KCODES + idx], 1u);
      out[OFF_IDX + (size_t)q * N_TOK + m] = (float)idx;
    }
  }

  s_red[tid] = acc;
  __syncthreads();
  for (int s = 128; s > 0; s >>= 1) {
    if (tid < s) s_red[tid] += s_red[tid + s];
    __syncthreads();
  }
  if (tid == 0) atomicAdd(&loss_acc[q], s_red[0]);
}

// ---------------------------------------------------------------------------
// quantized_out = sum over q of all_quantized[q]  (float2 granularity)
// ---------------------------------------------------------------------------
__global__ void rvq_sum(float* __restrict__ out) {
  size_t i = (size_t)blockIdx.x * blockDim.x + threadIdx.x;
  const size_t planes = (size_t)PLANE / 2;
  const float2* p = (const float2*)(out + OFF_ALLQ);
  float2 a0 = p[i];
  float2 a1 = p[i + planes];
  float2 a2 = p[i + 2 * planes];
  float2 a3 = p[i + 3 * planes];
  float2 r = make_float2(a0.x + a1.x + a2.x + a3.x,
                         a0.y + a1.y + a2.y + a3.y);
  ((float2*)(out + OFF_QOUT))[i] = r;
}

// ---------------------------------------------------------------------------
// total_loss and perplexity scalars
// ---------------------------------------------------------------------------
__global__ void rvq_finalize(const unsigned* __restrict__ hist,
                             const float* __restrict__ loss_acc,
                             float* __restrict__ out) {
  __shared__ float s_h[NQ][256];
  int tid = threadIdx.x;
  float h[NQ] = {0.f, 0.f, 0.f, 0.f};
  for (int k = tid; k < KCODES; k += 256) {
#pragma unroll
    for (int q = 0; q < NQ; ++q) {
      float pk = (float)hist[q * KCODES + k] * (1.0f / (float)N_TOK);
      h[q] += pk * logf(pk + 1e-10f);
    }
  }
#pragma unroll
  for (int q = 0; q < NQ; ++q) s_h[q][tid] = h[q];
  __syncthreads();
  for (int s = 128; s > 0; s >>= 1) {
    if (tid < s)
#pragma unroll
      for (int q = 0; q < NQ; ++q) s_h[q][tid] += s_h[q][tid + s];
    __syncthreads();
  }
  if (tid == 0) {
    float perp = 0.f, loss = 0.f;
    const float invND = 1.0f / ((float)N_TOK * (float)DIM);
#pragma unroll
    for (int q = 0; q < NQ; ++q) {
      perp += expf(-s_h[q][0]);
      loss += 1.25f * loss_acc[q] * invND;  // codebook + 0.25*commit (equal in value)
    }
    out[OFF_LOSS] = loss;
    out[OFF_PERP] = perp;
  }
}

// ---------------------------------------------------------------------------
extern "C" void kernel_launch(void* const* d_in, const int* in_sizes, int n_in,
                              void* d_out, int out_size, void* d_ws, size_t ws_size,
                              hipStream_t stream) {
  const float* x  = (const float*)d_in[0];   // [32,2048,256] f32
  const float* cb = (const float*)d_in[1];   // [4,1024,256] f32
  float* out = (float*)d_out;
  unsigned* hist    = (unsigned*)d_ws;                                  // NQ*KCODES u32
  float*   loss_acc = (float*)((char*)d_ws + (size_t)NQ * KCODES * 4);  // NQ f32
  float*   csq      = (float*)((char*)d_ws + (size_t)NQ * KCODES * 4 + 16); // NQ*KCODES f32

  rvq_init<<<(NQ * KCODES + NQ + 255) / 256, 256, 0, stream>>>((unsigned*)d_ws);

  rvq_csq<<<NQ * KCODES / 8, 256, 0, stream>>>(cb, csq);

  dim3 grid(N_TOK / ROWS_PER_BLOCK, NQ);
  rvq_argmin<<<grid, 256, 0, stream>>>(x, cb, csq, out, hist, loss_acc);

  rvq_sum<<<(PLANE / 2) / 256, 256, 0, stream>>>(out);

  rvq_finalize<<<1, 256, 0, stream>>>(hist, loss_acc, out);
}